// CompoundProteinInteractionPrediction_17995912970689
// MI455X (gfx1250) — compile-verified
//
#include <hip/hip_runtime.h>

typedef __attribute__((ext_vector_type(2))) float v2f;
typedef __attribute__((ext_vector_type(8))) float v8f;

#define N_ATOMS   8192
#define DIM       32
#define N_WORDS   4096
#define KSPLIT    4

// ---------------------------------------------------------------------------
// Embedding gather: out[g] = emb[idx[g/32]*32 + g%32]
// ---------------------------------------------------------------------------
__global__ void gather_emb(const int* __restrict__ idx, const float* __restrict__ emb,
                           float* __restrict__ out, int total) {
    int g = blockIdx.x * blockDim.x + threadIdx.x;
    if (g < total) out[g] = emb[(size_t)idx[g >> 5] * DIM + (g & 31)];
}

// ---------------------------------------------------------------------------
// hsT[n][m] = relu( sum_j xs[m][j] * W[j][n] + b[n] )  via V_WMMA_F32_16X16X4_F32
// One wave handles 16 rows x 32 cols (two 16-wide N tiles), K = 32 (8 steps).
// ---------------------------------------------------------------------------
__global__ void gnn_hs(const float* __restrict__ xs, const float* __restrict__ W,
                       const float* __restrict__ b, float* __restrict__ hsT) {
    const int lane = threadIdx.x & 31;
    const int wave = (blockIdx.x * blockDim.x + threadIdx.x) >> 5;   // 0..511
    const int lo = lane & 15;
    const int hi = lane >> 4;
    const int rowBase = wave * 16;

    const float* aPtr = xs + (size_t)(rowBase + lo) * DIM + 2 * hi;

    v8f acc0 = {}, acc1 = {};
#pragma unroll
    for (int k = 0; k < DIM; k += 4) {
        v2f a = *(const v2f*)(aPtr + k);
        int r = k + 2 * hi;
        v2f b0, b1;
        b0.x = W[r * DIM + lo];        b0.y = W[(r + 1) * DIM + lo];
        b1.x = W[r * DIM + 16 + lo];   b1.y = W[(r + 1) * DIM + 16 + lo];
        acc0 = __builtin_amdgcn_wmma_f32_16x16x4_f32(false, a, false, b0, (short)0, acc0, false, false);
        acc1 = __builtin_amdgcn_wmma_f32_16x16x4_f32(false, a, false, b1, (short)0, acc1, false, false);
    }

    const float bias0 = b[lo];
    const float bias1 = b[16 + lo];
#pragma unroll
    for (int i = 0; i < 8; ++i) {
        int m = rowBase + i + 8 * hi;                    // D layout: VGPR i, lane half
        hsT[(size_t)lo * N_ATOMS + m]        = fmaxf(acc0[i] + bias0, 0.f);
        hsT[(size_t)(16 + lo) * N_ATOMS + m] = fmaxf(acc1[i] + bias1, 0.f);
    }
}

// ---------------------------------------------------------------------------
// partial[chunk][m][n] = sum_{k in chunk} adj[m][k] * hsT[n][k]
// One wave: 16 rows x 32 cols, K chunk of 2048. b64 fragment loads for A and B
// (hsT transposed layout makes B loads contiguous along k, same as A).
// Adjacency uses non-temporal loads: 256 MB streamed, never reused.
// ---------------------------------------------------------------------------
__global__ void gnn_spmm(const float* __restrict__ adj, const float* __restrict__ hsT,
                         float* __restrict__ partial, int kChunk) {
    const int lane = threadIdx.x & 31;
    const int waveG = (blockIdx.x * blockDim.x + threadIdx.x) >> 5;  // 0..2047
    const int rowBlock = waveG >> 2;          // 0..511
    const int chunk    = waveG & (KSPLIT - 1);
    const int lo = lane & 15;
    const int hi = lane >> 4;
    const int rowBase = rowBlock * 16;
    const int k0 = chunk * kChunk;

    const float* aPtr  = adj + (size_t)(rowBase + lo) * N_ATOMS + k0 + 2 * hi;
    const float* b0Ptr = hsT + (size_t)lo * N_ATOMS        + k0 + 2 * hi;
    const float* b1Ptr = hsT + (size_t)(16 + lo) * N_ATOMS + k0 + 2 * hi;

    v8f acc0 = {}, acc1 = {};
#pragma unroll 4
    for (int k = 0; k < kChunk; k += 4) {
        v2f a  = __builtin_nontemporal_load((const v2f*)(aPtr + k));
        v2f b0 = *(const v2f*)(b0Ptr + k);
        v2f b1 = *(const v2f*)(b1Ptr + k);
        acc0 = __builtin_amdgcn_wmma_f32_16x16x4_f32(false, a, false, b0, (short)0, acc0, false, false);
        acc1 = __builtin_amdgcn_wmma_f32_16x16x4_f32(false, a, false, b1, (short)0, acc1, false, false);
    }

    float* out = partial + (size_t)chunk * (N_ATOMS * DIM);
#pragma unroll
    for (int i = 0; i < 8; ++i) {
        int m = rowBase + i + 8 * hi;
        out[(size_t)m * DIM + lo]      = acc0[i];
        out[(size_t)m * DIM + 16 + lo] = acc1[i];
    }
}

// ---------------------------------------------------------------------------
// out[g] = xs[g] + sum_c partial[c][g]   (residual + deterministic split-K sum)
// ---------------------------------------------------------------------------
__global__ void gnn_reduce(const float* __restrict__ xs, const float* __restrict__ p,
                           float* __restrict__ out) {
    int g = blockIdx.x * blockDim.x + threadIdx.x;
    float s = xs[g];
#pragma unroll
    for (int c = 0; c < KSPLIT; ++c) s += p[(size_t)c * (N_ATOMS * DIM) + g];
    out[g] = s;
}

// ---------------------------------------------------------------------------
extern "C" void kernel_launch(void* const* d_in, const int* in_sizes, int n_in,
                              void* d_out, int out_size, void* d_ws, size_t ws_size,
                              hipStream_t stream) {
    const int*   fp     = (const int*)  d_in[0];
    const float* adj    = (const float*)d_in[1];
    const int*   words  = (const int*)  d_in[2];
    const float* emb_fp = (const float*)d_in[3];
    const float* emb_w  = (const float*)d_in[4];
    const float* Wg     = (const float*)d_in[5];
    const float* bg     = (const float*)d_in[6];
    float* out = (float*)d_out;

    char* ws = (char*)d_ws;
    float* xsA  = (float*)(ws);
    float* xsB  = (float*)(ws + (size_t)(1u << 20));
    float* hsT  = (float*)(ws + (size_t)(2u << 20));
    float* part = (float*)(ws + (size_t)(3u << 20));   // KSPLIT * 1 MB

    const int NX = N_ATOMS * DIM;   // 262144 compound elements
    const int NW = N_WORDS * DIM;   // 131072 word elements

    gather_emb<<<NX / 256, 256, 0, stream>>>(fp, emb_fp, xsA, NX);
    gather_emb<<<NW / 256, 256, 0, stream>>>(words, emb_w, out + NX, NW);

    float* cur = xsA;
    float* nxt = xsB;
    for (int l = 0; l < 3; ++l) {
        gnn_hs  <<<64, 256, 0, stream>>>(cur, Wg + l * DIM * DIM, bg + l * DIM, hsT);
        gnn_spmm<<<256, 256, 0, stream>>>(adj, hsT, part, N_ATOMS / KSPLIT);
        float* dst = (l == 2) ? out : nxt;
        gnn_reduce<<<NX / 256, 256, 0, stream>>>(cur, part, dst);
        float* t = cur; cur = nxt; nxt = t;
    }
}